// Voxelization_70153995813272
// MI455X (gfx1250) — compile-verified
//
#include <hip/hip_runtime.h>
#include <stdint.h>

#define B_   8
#define C_   128
#define N_   100000
#define R_   32
#define R3_  32768
#define VOX_ELEMS (B_ * C_ * R3_)   // 33,554,432 floats

// workspace layout (floats): [0..31] stats (4 per batch), [32..] counts B*R3
#define WS_STATS 0
#define WS_CNT   32

// Native CDNA5 f32 global atomic add, device scope (no return -> STOREcnt).
__device__ __forceinline__ void atom_add_f32(float* p, float v) {
  asm volatile("global_atomic_add_f32 %0, %1, off scope:SCOPE_DEV"
               :: "v"((unsigned long long)(uintptr_t)p), "v"(v) : "memory");
}

// ---------------------------------------------------------------------------
// Feature scatter (placed first so the disasm snippet shows it).
// Block = 256 threads = 1024 points of batch b, half the channels (blockIdx.z).
// Coalesced float4 feature loads are staged through LDS with CDNA5 async
// loads (double-buffered; each thread reads back only its own slot, so
// per-wave s_wait_asynccnt ordering suffices -> no barrier), overlapped with
// the previous channel's scattered atomics into the L2-resident voxel slab.
// ---------------------------------------------------------------------------
#define PTS_PER_BLOCK 1024
#define CH_SPLIT 2
#define CH_PER (C_ / CH_SPLIT)

__global__ __launch_bounds__(256)
void k_scatter(const float* __restrict__ feat, const float* __restrict__ coords,
               const float* __restrict__ stats, float* __restrict__ vox) {
  __shared__ float4 fbuf[2][256];
  const int t  = threadIdx.x;
  const int b  = blockIdx.y;
  const int c0 = blockIdx.z * CH_PER;
  const int n0 = blockIdx.x * PTS_PER_BLOCK;
  const int n  = n0 + t * 4;
  const bool active = (n < N_);
  const int nc = active ? n : 0;   // clamped, stays 16B-aligned & in-bounds

  // recompute voxel indices for my 4 points
  const float* cb = coords + (size_t)b * 3 * N_;
  const float mx  = stats[b * 4 + 0];
  const float my  = stats[b * 4 + 1];
  const float mz  = stats[b * 4 + 2];
  const float inv = 0.5f / stats[b * 4 + 3];
  int idx[4];
#pragma unroll
  for (int k = 0; k < 4; ++k) {
    float x = (cb[nc + k]          - mx) * inv + 0.5f;
    float y = (cb[N_ + nc + k]     - my) * inv + 0.5f;
    float z = (cb[2 * N_ + nc + k] - mz) * inv + 0.5f;
    x = fminf(fmaxf(x * (float)R_, 0.f), (float)(R_ - 1));
    y = fminf(fmaxf(y * (float)R_, 0.f), (float)(R_ - 1));
    z = fminf(fmaxf(z * (float)R_, 0.f), (float)(R_ - 1));
    idx[k] = ((int)rintf(x) << 10) | ((int)rintf(y) << 5) | (int)rintf(z);
  }

  const float4* fb = (const float4*)(feat + (size_t)b * C_ * N_);
  const uint32_t lds0 = (uint32_t)(uintptr_t)&fbuf[0][t];
  const uint32_t lds1 = (uint32_t)(uintptr_t)&fbuf[1][t];

  // prologue: async-stage first channel of this block's range
  {
    unsigned long long ga =
        (unsigned long long)(uintptr_t)(fb + ((c0 * N_ + nc) >> 2));
    asm volatile("global_load_async_to_lds_b128 %0, %1, off"
                 :: "v"(lds0), "v"(ga) : "memory");
  }

  float* vb = vox + (((size_t)b * C_) << 15);
  for (int i = 0; i < CH_PER; ++i) {
    const int c = c0 + i;
    if (i + 1 < CH_PER) {
      unsigned long long ga =
          (unsigned long long)(uintptr_t)(fb + (((c + 1) * N_ + nc) >> 2));
      asm volatile("global_load_async_to_lds_b128 %0, %1, off"
                   :: "v"(((i + 1) & 1) ? lds1 : lds0), "v"(ga) : "memory");
      asm volatile("s_wait_asynccnt 0x1" ::: "memory"); // buf[i&1] ready
    } else {
      asm volatile("s_wait_asynccnt 0x0" ::: "memory");
    }
    const float4 f = fbuf[i & 1][t];
    if (active) {
      float* vc = vb + ((size_t)c << 15);
      atom_add_f32(vc + idx[0], f.x);
      atom_add_f32(vc + idx[1], f.y);
      atom_add_f32(vc + idx[2], f.z);
      atom_add_f32(vc + idx[3], f.w);
    }
  }
}

// ---------------------------------------------------------------------------
// Per-batch mean (3) + max L2 norm (scale). One 1024-thread block per batch.
// ---------------------------------------------------------------------------
__global__ __launch_bounds__(1024)
void k_stats(const float* __restrict__ coords, float* __restrict__ stats) {
  __shared__ float sm0[1024];
  __shared__ float sm1[1024];
  __shared__ float sm2[1024];
  const int b = blockIdx.x;
  const int t = threadIdx.x;
  const float* cb = coords + (size_t)b * 3 * N_;

  float sx = 0.f, sy = 0.f, sz = 0.f;
  for (int n = t; n < N_; n += 1024) {
    sx += cb[n];
    sy += cb[N_ + n];
    sz += cb[2 * N_ + n];
  }
  sm0[t] = sx; sm1[t] = sy; sm2[t] = sz;
  __syncthreads();
  for (int s = 512; s > 0; s >>= 1) {
    if (t < s) { sm0[t] += sm0[t + s]; sm1[t] += sm1[t + s]; sm2[t] += sm2[t + s]; }
    __syncthreads();
  }
  const float mx = sm0[0] * (1.0f / N_);
  const float my = sm1[0] * (1.0f / N_);
  const float mz = sm2[0] * (1.0f / N_);
  __syncthreads();   // all threads read sm*[0] before reuse

  float mq = 0.f;
  for (int n = t; n < N_; n += 1024) {
    const float x = cb[n] - mx;
    const float y = cb[N_ + n] - my;
    const float z = cb[2 * N_ + n] - mz;
    mq = fmaxf(mq, x * x + y * y + z * z);
  }
  sm0[t] = mq;
  __syncthreads();
  for (int s = 512; s > 0; s >>= 1) {
    if (t < s) sm0[t] = fmaxf(sm0[t], sm0[t + s]);
    __syncthreads();
  }
  if (t == 0) {
    stats[b * 4 + 0] = mx;
    stats[b * 4 + 1] = my;
    stats[b * 4 + 2] = mz;
    stats[b * 4 + 3] = sqrtf(sm0[0]);   // max ||nc||
  }
}

// ---------------------------------------------------------------------------
// Zero a float4 range (vox accumulator, counts)
// ---------------------------------------------------------------------------
__global__ __launch_bounds__(256)
void k_zero(float4* __restrict__ p, int n4) {
  int i = blockIdx.x * blockDim.x + threadIdx.x;
  const int stride = gridDim.x * blockDim.x;
  const float4 z = make_float4(0.f, 0.f, 0.f, 0.f);
  for (; i < n4; i += stride) p[i] = z;
}

// ---------------------------------------------------------------------------
// norm_coords output + voxel count accumulation
// ---------------------------------------------------------------------------
__global__ __launch_bounds__(256)
void k_coords(const float* __restrict__ coords, const float* __restrict__ stats,
              float* __restrict__ ncoords, float* __restrict__ cnt) {
  const int n = blockIdx.x * 256 + threadIdx.x;
  const int b = blockIdx.y;
  if (n >= N_) return;
  const float* cb = coords + (size_t)b * 3 * N_;
  const float mx  = stats[b * 4 + 0];
  const float my  = stats[b * 4 + 1];
  const float mz  = stats[b * 4 + 2];
  const float inv = 0.5f / stats[b * 4 + 3];   // 1/(2*scale)

  float x = (cb[n]          - mx) * inv + 0.5f;
  float y = (cb[N_ + n]     - my) * inv + 0.5f;
  float z = (cb[2 * N_ + n] - mz) * inv + 0.5f;
  x = fminf(fmaxf(x * (float)R_, 0.f), (float)(R_ - 1));
  y = fminf(fmaxf(y * (float)R_, 0.f), (float)(R_ - 1));
  z = fminf(fmaxf(z * (float)R_, 0.f), (float)(R_ - 1));

  float* nb = ncoords + (size_t)b * 3 * N_;
  nb[n]          = x;
  nb[N_ + n]     = y;
  nb[2 * N_ + n] = z;

  const int ix = (int)rintf(x);   // RNE, matches jnp.round
  const int iy = (int)rintf(y);
  const int iz = (int)rintf(z);
  const int flat = (ix << 10) | (iy << 5) | iz;
  atom_add_f32(&cnt[b * R3_ + flat], 1.0f);
}

// ---------------------------------------------------------------------------
// Divide by max(cnt, 1): one thread owns 4 consecutive voxels (float4 RMW),
// reuses its 4 reciprocals across all 128 channels.
// ---------------------------------------------------------------------------
__global__ __launch_bounds__(256)
void k_norm(float4* __restrict__ vox, const float4* __restrict__ cnt) {
  const int v4 = blockIdx.x * 256 + threadIdx.x;  // 0 .. R3/4-1
  const int b  = blockIdx.y;
  const float4 c = cnt[b * (R3_ / 4) + v4];
  float4 inv;
  inv.x = 1.0f / fmaxf(c.x, 1.0f);
  inv.y = 1.0f / fmaxf(c.y, 1.0f);
  inv.z = 1.0f / fmaxf(c.z, 1.0f);
  inv.w = 1.0f / fmaxf(c.w, 1.0f);
  float4* p = vox + (size_t)b * C_ * (R3_ / 4) + v4;
#pragma unroll 4
  for (int ch = 0; ch < C_; ++ch) {
    float4 f = p[(size_t)ch * (R3_ / 4)];
    f.x *= inv.x; f.y *= inv.y; f.z *= inv.z; f.w *= inv.w;
    p[(size_t)ch * (R3_ / 4)] = f;
  }
}

// ---------------------------------------------------------------------------
extern "C" void kernel_launch(void* const* d_in, const int* in_sizes, int n_in,
                              void* d_out, int out_size, void* d_ws, size_t ws_size,
                              hipStream_t stream) {
  (void)in_sizes; (void)n_in; (void)out_size; (void)ws_size;
  const float* feat   = (const float*)d_in[0];   // [8,128,100000]
  const float* coords = (const float*)d_in[1];   // [8,3,100000]
  float* vox     = (float*)d_out;                // [8,128,32768]
  float* ncoords = (float*)d_out + VOX_ELEMS;    // [8,3,100000]
  float* stats   = (float*)d_ws + WS_STATS;      // 8*4
  float* cnt     = (float*)d_ws + WS_CNT;        // 8*32768

  k_stats<<<B_, 1024, 0, stream>>>(coords, stats);
  k_zero<<<2048, 256, 0, stream>>>((float4*)vox, VOX_ELEMS / 4);
  k_zero<<<256, 256, 0, stream>>>((float4*)cnt, (B_ * R3_) / 4);

  dim3 g3((N_ + 255) / 256, B_);
  k_coords<<<g3, 256, 0, stream>>>(coords, stats, ncoords, cnt);

  dim3 g4((N_ + PTS_PER_BLOCK - 1) / PTS_PER_BLOCK, B_, CH_SPLIT);
  k_scatter<<<g4, 256, 0, stream>>>(feat, coords, stats, vox);

  dim3 g5(R3_ / 4 / 256, B_);
  k_norm<<<g5, 256, 0, stream>>>((float4*)vox, (const float4*)cnt);
}